// GINBaseline_11819749999035
// MI455X (gfx1250) — compile-verified
//
#include <hip/hip_runtime.h>
#include <hip/hip_bf16.h>

typedef float v2f __attribute__((ext_vector_type(2)));
typedef float v8f __attribute__((ext_vector_type(8)));

#define GIN_B 8
#define GIN_N 10000
#define GIN_E 1280000
#define GIN_D 128
#define GIN_L 4
#define GIN_TOTAL (GIN_B * GIN_N)   // 80000 nodes
#define LN_EPS 1e-5f

// ---------------------------------------------------------------------------
// h[i][:] = emb[x[i]][:]   (one wave per node, float4 per lane)
// ---------------------------------------------------------------------------
__global__ __launch_bounds__(256) void gin_gather_emb(
    const int* __restrict__ x, const float* __restrict__ emb,
    float* __restrict__ h)
{
    int wid  = (blockIdx.x * blockDim.x + threadIdx.x) >> 5;
    int lane = threadIdx.x & 31;
    if (wid >= GIN_TOTAL) return;
    int v = x[wid];
    const float4 val = *(const float4*)&emb[(long long)v * GIN_D + lane * 4];
    *(float4*)&h[(long long)wid * GIN_D + lane * 4] = val;
}

// ---------------------------------------------------------------------------
// zero the pooled accumulator [L][B][D]
// ---------------------------------------------------------------------------
__global__ __launch_bounds__(256) void gin_zero_pooled(float* __restrict__ pooled)
{
    int i = blockIdx.x * blockDim.x + threadIdx.x;
    if (i < GIN_L * GIN_B * GIN_D) pooled[i] = 0.0f;
}

// ---------------------------------------------------------------------------
// agg[i][:] = (1 + eps[layer]) * h[i][:]    (float4 per thread)
// ---------------------------------------------------------------------------
__global__ __launch_bounds__(256) void gin_init_agg(
    const float* __restrict__ h, float* __restrict__ agg,
    const float* __restrict__ eps, int layer)
{
    float scale = 1.0f + eps[layer];
    long long i = (long long)blockIdx.x * blockDim.x + threadIdx.x;   // float4 index
    if (i >= (long long)GIN_TOTAL * GIN_D / 4) return;
    float4 v = ((const float4*)h)[i];
    v.x *= scale; v.y *= scale; v.z *= scale; v.w *= scale;
    ((float4*)agg)[i] = v;
}

// ---------------------------------------------------------------------------
// agg[dst[e]][:] += h[src[e]][:]   (one wave per edge; 4 f32 atomics per lane)
// h and agg are L2-resident (41 MB each, 192 MB L2) -> atomics resolve in L2.
// ---------------------------------------------------------------------------
__global__ __launch_bounds__(256) void gin_scatter_add(
    const float* __restrict__ h, float* __restrict__ agg,
    const int* __restrict__ src, const int* __restrict__ dst)
{
    int wid  = (blockIdx.x * blockDim.x + threadIdx.x) >> 5;
    int lane = threadIdx.x & 31;
    if (wid >= GIN_E) return;
    int s = src[wid];
    int d = dst[wid];
    const float4 v = *(const float4*)&h[(long long)s * GIN_D + lane * 4];
    float* p = &agg[(long long)d * GIN_D + lane * 4];
    atomicAdd(p + 0, v.x);
    atomicAdd(p + 1, v.y);
    atomicAdd(p + 2, v.z);
    atomicAdd(p + 3, v.w);
}

// ---------------------------------------------------------------------------
// Fused MLP: Hout = relu(LN(X @ Wa + ba) * g + be) @ Wb + bb,
// plus per-batch sum-pool accumulation into pooled_layer[b][:].
//
// Block = 256 threads = 8 waves, owns a 16-row tile of X (10000 % 16 == 0 so a
// tile never straddles a batch). Wave w computes output columns [16w,16w+16)
// via V_WMMA_F32_16X16X4_F32, accumulating K=128 in 32 WMMAs per GEMM.
//
// f32 WMMA fragment layouts (ISA 7.12.2):
//   A 16x4:  lanes 0-15: M=lane,   VGPR0=K0, VGPR1=K1 ; lanes 16-31: K2,K3
//   B 4x16:  lanes 0-15: N=lane,   VGPR0=K0, VGPR1=K1 ; lanes 16-31: K2,K3
//   C 16x16: VGPR v: lanes 0-15 -> M=v, N=lane ; lanes 16-31 -> M=v+8
// ---------------------------------------------------------------------------
__global__ __launch_bounds__(256) void gin_mlp_wmma(
    const float* __restrict__ X, float* __restrict__ Hout,
    const float* __restrict__ Wa, const float* __restrict__ ba,
    const float* __restrict__ g,  const float* __restrict__ be,
    const float* __restrict__ Wb, const float* __restrict__ bb,
    float* __restrict__ pooled_layer)
{
    __shared__ float Xs[16][132];   // padded stride to dodge bank conflicts
    __shared__ float Hs[16][132];
    __shared__ float mu_s[16], rs_s[16];

    const int tid  = threadIdx.x;
    const int w    = tid >> 5;        // wave 0..7 -> column tile
    const int lane = tid & 31;
    const int row0 = blockIdx.x * 16;
    const int b    = row0 / GIN_N;    // batch of this tile

    // --- load 16x128 input tile (float4 per thread) -------------------------
    for (int i = tid; i < 16 * 32; i += 256) {
        int r  = i >> 5;
        int c4 = (i & 31) * 4;
        const float4 v = *(const float4*)&X[(long long)(row0 + r) * GIN_D + c4];
        Xs[r][c4 + 0] = v.x; Xs[r][c4 + 1] = v.y;
        Xs[r][c4 + 2] = v.z; Xs[r][c4 + 3] = v.w;
    }
    __syncthreads();

    const int n       = lane & 15;          // A row (M) and B column (N)
    const int khalf   = (lane >> 4) * 2;    // 0 for lanes 0-15, 2 for 16-31
    const int colbase = w * 16;
    const int mbase   = (lane >> 4) * 8;    // C/D row base per lane half

    // --- GEMM 1: C = Xtile @ Wa ---------------------------------------------
    v8f c = {};
#pragma unroll
    for (int k = 0; k < GIN_D; k += 4) {
        v2f a, bf;
        a.x  = Xs[n][k + khalf];
        a.y  = Xs[n][k + khalf + 1];
        bf.x = Wa[(k + khalf)     * GIN_D + colbase + n];
        bf.y = Wa[(k + khalf + 1) * GIN_D + colbase + n];
        c = __builtin_amdgcn_wmma_f32_16x16x4_f32(
                false, a, false, bf, (short)0, c, false, false);
    }
    {
        float bias = ba[colbase + n];
#pragma unroll
        for (int v = 0; v < 8; ++v)
            Hs[mbase + v][colbase + n] = c[v] + bias;
    }
    __syncthreads();

    // --- LayerNorm statistics (one thread per row) --------------------------
    if (tid < 16) {
        float s = 0.0f, s2 = 0.0f;
        for (int k = 0; k < GIN_D; ++k) {
            float x = Hs[tid][k];
            s += x; s2 += x * x;
        }
        float mu  = s * (1.0f / GIN_D);
        float var = s2 * (1.0f / GIN_D) - mu * mu;
        mu_s[tid] = mu;
        rs_s[tid] = rsqrtf(var + LN_EPS);
    }
    __syncthreads();

    // --- normalize * g + be, ReLU -> Xs -------------------------------------
    for (int i = tid; i < 16 * GIN_D; i += 256) {
        int r  = i >> 7;
        int cc = i & 127;
        float x = (Hs[r][cc] - mu_s[r]) * rs_s[r] * g[cc] + be[cc];
        Xs[r][cc] = x > 0.0f ? x : 0.0f;
    }
    __syncthreads();

    // --- GEMM 2: D = relu(LN(...)) @ Wb -------------------------------------
    v8f c2 = {};
#pragma unroll
    for (int k = 0; k < GIN_D; k += 4) {
        v2f a, bf;
        a.x  = Xs[n][k + khalf];
        a.y  = Xs[n][k + khalf + 1];
        bf.x = Wb[(k + khalf)     * GIN_D + colbase + n];
        bf.y = Wb[(k + khalf + 1) * GIN_D + colbase + n];
        c2 = __builtin_amdgcn_wmma_f32_16x16x4_f32(
                false, a, false, bf, (short)0, c2, false, false);
    }
    {
        float bias = bb[colbase + n];
#pragma unroll
        for (int v = 0; v < 8; ++v) {
            float val = c2[v] + bias;
            Hout[(long long)(row0 + mbase + v) * GIN_D + colbase + n] = val;
            Hs[mbase + v][colbase + n] = val;   // keep for pooling
        }
    }
    __syncthreads();

    // --- sum-pool: reduce 16 rows in-block, then 128 atomics ----------------
    if (tid < GIN_D) {
        float s = 0.0f;
#pragma unroll
        for (int r = 0; r < 16; ++r) s += Hs[r][tid];
        atomicAdd(&pooled_layer[b * GIN_D + tid], s);
    }
}

// ---------------------------------------------------------------------------
// Classifier: out[b] = relu(ge[b] @ wc1 + bc1) @ wc2 + bc2
// ge[b] = concat_i pooled[i][b][:]   -> [8][512]; hidden [8][1024]. One block.
// ---------------------------------------------------------------------------
__global__ __launch_bounds__(256) void gin_classifier(
    const float* __restrict__ pooled,   // [L][B][D]
    const float* __restrict__ wc1,      // [512][1024]
    const float* __restrict__ bc1,      // [1024]
    const float* __restrict__ wc2,      // [1024]
    const float* __restrict__ bc2,      // [1]
    float* __restrict__ out)            // [8]
{
    __shared__ float ge[GIN_B][GIN_L * GIN_D];   // 8x512 = 16 KB
    __shared__ float red[256][GIN_B];            // 8 KB

    const int tid = threadIdx.x;
    const int LD  = GIN_L * GIN_D;               // 512
    const int H   = 2 * GIN_L * GIN_D;           // 1024

    for (int i = tid; i < GIN_B * LD; i += 256) {
        int b = i / LD;
        int k = i - b * LD;
        int layer = k >> 7;
        int d     = k & 127;
        ge[b][k] = pooled[(layer * GIN_B + b) * GIN_D + d];
    }
    __syncthreads();

    float part[GIN_B];
#pragma unroll
    for (int b = 0; b < GIN_B; ++b) part[b] = 0.0f;

    for (int j = tid; j < H; j += 256) {
        float w2 = wc2[j];
        float b1 = bc1[j];
#pragma unroll
        for (int b = 0; b < GIN_B; ++b) {
            float acc = b1;
            for (int k = 0; k < LD; ++k)
                acc += ge[b][k] * wc1[k * H + j];
            acc = acc > 0.0f ? acc : 0.0f;
            part[b] += acc * w2;
        }
    }
#pragma unroll
    for (int b = 0; b < GIN_B; ++b) red[tid][b] = part[b];
    __syncthreads();

    if (tid < GIN_B) {
        float s = 0.0f;
        for (int t = 0; t < 256; ++t) s += red[t][tid];
        out[tid] = s + bc2[0];
    }
}

// ---------------------------------------------------------------------------
extern "C" void kernel_launch(void* const* d_in, const int* in_sizes, int n_in,
                              void* d_out, int out_size, void* d_ws, size_t ws_size,
                              hipStream_t stream)
{
    const int*   x    = (const int*)  d_in[0];
    const int*   ei   = (const int*)  d_in[1];     // [2][E]
    const float* emb  = (const float*)d_in[2];
    const float* w1a  = (const float*)d_in[3];
    const float* b1a  = (const float*)d_in[4];
    const float* g1   = (const float*)d_in[5];
    const float* be1  = (const float*)d_in[6];
    const float* w1b  = (const float*)d_in[7];
    const float* b1b  = (const float*)d_in[8];
    const float* wha  = (const float*)d_in[9];
    const float* bha  = (const float*)d_in[10];
    const float* gh   = (const float*)d_in[11];
    const float* beh  = (const float*)d_in[12];
    const float* whb  = (const float*)d_in[13];
    const float* bhb  = (const float*)d_in[14];
    const float* eps  = (const float*)d_in[15];
    const float* wc1  = (const float*)d_in[16];
    const float* bc1  = (const float*)d_in[17];
    const float* wc2  = (const float*)d_in[18];
    const float* bc2  = (const float*)d_in[19];

    const int* src = ei;
    const int* dst = ei + GIN_E;

    // Workspace layout (all L2-resident on MI455X: 192 MB L2)
    char* ws = (char*)d_ws;
    float* h      = (float*)(ws);                                          // 41 MB
    float* agg    = (float*)(ws + (size_t)GIN_TOTAL * GIN_D * 4);          // 41 MB
    float* pooled = (float*)(ws + (size_t)GIN_TOTAL * GIN_D * 4 * 2);      // 16 KB

    // 1) embedding gather: one wave per node
    gin_gather_emb<<<(GIN_TOTAL * 32 + 255) / 256, 256, 0, stream>>>(x, emb, h);

    // 2) zero pooled accumulator
    gin_zero_pooled<<<(GIN_L * GIN_B * GIN_D + 255) / 256, 256, 0, stream>>>(pooled);

    // 3) GIN layers
    const long long nvec4 = (long long)GIN_TOTAL * GIN_D / 4;
    for (int layer = 0; layer < GIN_L; ++layer) {
        const float* Wa = (layer == 0) ? w1a : wha;
        const float* Ba = (layer == 0) ? b1a : bha;
        const float* Gg = (layer == 0) ? g1  : gh;
        const float* Be = (layer == 0) ? be1 : beh;
        const float* Wb = (layer == 0) ? w1b : whb;
        const float* Bb = (layer == 0) ? b1b : bhb;

        gin_init_agg<<<(int)((nvec4 + 255) / 256), 256, 0, stream>>>(h, agg, eps, layer);
        gin_scatter_add<<<(GIN_E * 32) / 256, 256, 0, stream>>>(h, agg, src, dst);
        gin_mlp_wmma<<<GIN_TOTAL / 16, 256, 0, stream>>>(
            agg, h, Wa, Ba, Gg, Be, Wb, Bb, pooled + layer * GIN_B * GIN_D);
    }

    // 4) classifier head
    gin_classifier<<<1, 256, 0, stream>>>(pooled, wc1, bc1, wc2, bc2, (float*)d_out);
}